// BioREDirect_64527588655498
// MI455X (gfx1250) — compile-verified
//
#include <hip/hip_runtime.h>
#include <cmath>

// Problem constants (from the reference)
#define Bn 64
#define Sn 512
#define Hn 768
#define Kn 64

typedef __attribute__((ext_vector_type(2))) float v2f;
typedef __attribute__((ext_vector_type(8))) float v8f;

// Pt is stored k-pair interleaved: Ptp[((h>>1)*Bn + b)*2 + (h&1)]
// so a WMMA B-fragment lane (needs Pt[k][b], Pt[k+1][b]) is one aligned float2.
__device__ __forceinline__ size_t ptp_index(int h, int b) {
    return ((size_t)(h >> 1) * Bn + b) * 2 + (h & 1);
}

// -------------------------------------------------------------------------
// Kernel 1: ragged intra-group attention pooling.
// One 256-thread block per (task t, batch b). Writes Ptp[t][...] (pair-
// interleaved transpose) so kernel-2 B-fragment loads are single b64, lane-
// coalesced.
// -------------------------------------------------------------------------
struct PoolArgs {
    const float* seq;        // [B,S,H]
    const int*   idx[3];     // [B,K]
    const int*   len[3];     // [B]
    float*       Pt;         // [3][H/2][B][2]
};

__global__ __launch_bounds__(256) void pool_kernel(PoolArgs args) {
    const int blk  = blockIdx.x;         // 0..191
    const int t    = blk / Bn;
    const int b    = blk % Bn;
    const int tid  = threadIdx.x;
    const int lane = tid & 31;
    const int wave = tid >> 5;

    __shared__ int   s_idx[Kn];
    __shared__ float s_g0[Hn];
    __shared__ float s_sc[Kn];
    __shared__ int   s_len;

    const int* idx = args.idx[t] + b * Kn;
    if (tid < Kn) s_idx[tid] = idx[tid];
    if (tid == 0) s_len = args.len[t][b];
    __syncthreads();

    const int len = s_len;
    const int i0  = s_idx[0];
    const float* seqb = args.seq + (size_t)b * Sn * Hn;

    // stage G0 = seq[b, idx[0], :] in LDS (coalesced)
    for (int h = tid; h < Hn; h += 256) s_g0[h] = seqb[(size_t)i0 * Hn + h];
    __syncthreads();

    // scores0[j] = dot(G0, G_j); wave w handles j = w, w+8, ...
    for (int j = wave; j < Kn; j += 8) {
        const float* row = seqb + (size_t)s_idx[j] * Hn;
        float acc = 0.f;
        for (int h = lane; h < Hn; h += 32) acc += s_g0[h] * row[h];
        for (int o = 16; o > 0; o >>= 1) acc += __shfl_xor(acc, o, 32);
        if (lane == 0) s_sc[j] = (j < len) ? acc : -1e30f;
    }
    __syncthreads();

    // softmax over 64 scores (computed redundantly; all threads agree)
    float m = -1e30f;
    for (int j = 0; j < Kn; ++j) m = fmaxf(m, s_sc[j]);
    float ssum = 0.f;
    for (int j = 0; j < Kn; ++j) ssum += expf(s_sc[j] - m);
    const float inv = 1.0f / ssum;
    __syncthreads();
    if (tid < Kn) s_sc[tid] = expf(s_sc[tid] - m) * inv;   // a0[j]
    __syncthreads();

    // pooled[h] = sum_j a0[j] * seq[b, idx[j], h]  (coalesced over h)
    float* Ptp = args.Pt + (size_t)t * Hn * Bn;
    for (int h = tid; h < Hn; h += 256) {
        float acc = 0.f;
        for (int j = 0; j < Kn; ++j)
            acc += s_sc[j] * seqb[(size_t)s_idx[j] * Hn + h];
        if (len == 0) acc = seqb[h];     // fallback: seq[b, 0, h]
        Ptp[ptp_index(h, b)] = acc;
    }
}

// -------------------------------------------------------------------------
// Kernel 2: Hdn[t][b][o] = tanh( sum_k Wd_t[o,k] * Pt[t][k][b] + bd_t[o] )
// via V_WMMA_F32_16X16X4_F32.
// One wave per (task, 16-row o-tile): keeps ONE A fragment (Wd rows,
// contiguous float2 per lane) and accumulates FOUR 16x16 tiles covering all
// 64 batches -> 4 WMMAs per A load, Wd read exactly once grid-wide.
// Software-pipelined: next step's 5 fragments are loaded before this step's
// WMMAs so the wait doesn't drain loadcnt to 0.
// -------------------------------------------------------------------------
struct GemmArgs {
    const float* Wd[3];      // [H,H] row-major (o, k)
    const float* bd[3];      // [H]
    const float* Pt;         // [3][H/2][B][2]
    float*       Hdn;        // [3][B][H]
};

__global__ __launch_bounds__(256) void dense_tanh_wmma(GemmArgs args) {
    const int tid  = threadIdx.x;
    const int lane = tid & 31;
    const int wv   = blockIdx.x * 8 + (tid >> 5);    // 0..143
    const int t     = wv / 48;
    const int otile = wv % 48;

    const float* Wd  = args.Wd[t];
    const float* Ptp = args.Pt + (size_t)t * Hn * Bn;

    const int mrow = otile * 16 + (lane & 15);       // o for A fragment
    const int nl   = lane & 15;                      // b within tile
    const int ksel = (lane >> 4) << 1;               // 0 or 2 (K split by half-wave)

    const float* ap = Wd + (size_t)mrow * Hn;

    v8f c0 = {}, c1 = {}, c2 = {}, c3 = {};

    // prologue loads for k0 = 0
    v2f a  = *(const v2f*)(ap + ksel);
    v2f b0 = *(const v2f*)(Ptp + ptp_index(ksel,  0 + nl));
    v2f b1 = *(const v2f*)(Ptp + ptp_index(ksel, 16 + nl));
    v2f b2 = *(const v2f*)(Ptp + ptp_index(ksel, 32 + nl));
    v2f b3 = *(const v2f*)(Ptp + ptp_index(ksel, 48 + nl));

    for (int k0 = 0; k0 < Hn; k0 += 4) {
        const v2f ac = a, bc0 = b0, bc1 = b1, bc2 = b2, bc3 = b3;
        const int kn = k0 + 4 + ksel;
        if (k0 + 4 < Hn) {                           // preload next step
            a  = *(const v2f*)(ap + kn);
            b0 = *(const v2f*)(Ptp + ptp_index(kn,  0 + nl));
            b1 = *(const v2f*)(Ptp + ptp_index(kn, 16 + nl));
            b2 = *(const v2f*)(Ptp + ptp_index(kn, 32 + nl));
            b3 = *(const v2f*)(Ptp + ptp_index(kn, 48 + nl));
            __builtin_prefetch(ap + kn + 64, 0, 1);  // global_prefetch_b8 ahead on Wd row
        }
        c0 = __builtin_amdgcn_wmma_f32_16x16x4_f32(false, ac, false, bc0, (short)0, c0, false, false);
        c1 = __builtin_amdgcn_wmma_f32_16x16x4_f32(false, ac, false, bc1, (short)0, c1, false, false);
        c2 = __builtin_amdgcn_wmma_f32_16x16x4_f32(false, ac, false, bc2, (short)0, c2, false, false);
        c3 = __builtin_amdgcn_wmma_f32_16x16x4_f32(false, ac, false, bc3, (short)0, c3, false, false);
    }

    // Epilogue: bias + tanh; C/D layout: VGPR v -> M=v (lanes 0-15), M=v+8 (16-31)
    const float* bd  = args.bd[t];
    float*       Hdn = args.Hdn + (size_t)t * Bn * Hn;
    const int mo = ((lane >> 4) << 3);
    v8f cs[4] = { c0, c1, c2, c3 };
    for (int i = 0; i < 4; ++i) {
        const int bb = i * 16 + nl;
        for (int v = 0; v < 8; ++v) {
            const int oo = otile * 16 + v + mo;
            Hdn[(size_t)bb * Hn + oo] = tanhf(cs[i][v] + bd[oo]);
        }
    }
}

// -------------------------------------------------------------------------
// Kernel 3: logits[t][b][c] = sum_o Hdn[t][b][o] * Wc_t[c,o] + bc_t[c]
// One wave per (t,b); nlab <= 9.
// -------------------------------------------------------------------------
struct ClsArgs {
    const float* Hdn;        // [3][B][H]
    const float* Wc[3];
    const float* bc[3];
    float*       out;        // 960 floats, concat (rel 64x9, nov 64x3, dir 64x3)
    int          nlab[3];
    int          ooff[3];
};

__global__ __launch_bounds__(256) void cls_kernel(ClsArgs args) {
    const int tid  = threadIdx.x;
    const int lane = tid & 31;
    const int wv   = blockIdx.x * 8 + (tid >> 5);    // 0..191
    const int t = wv / Bn;
    const int b = wv % Bn;
    const int nlab = args.nlab[t];

    const float* hrow = args.Hdn + ((size_t)t * Bn + b) * Hn;
    const float* Wc   = args.Wc[t];

    float acc[9];
    for (int c = 0; c < 9; ++c) acc[c] = 0.f;
    for (int o = lane; o < Hn; o += 32) {
        const float hv = hrow[o];
        for (int c = 0; c < nlab; ++c) acc[c] += hv * Wc[(size_t)c * Hn + o];
    }
    for (int c = 0; c < nlab; ++c)
        for (int off = 16; off > 0; off >>= 1)
            acc[c] += __shfl_xor(acc[c], off, 32);
    if (lane == 0) {
        const float* bc = args.bc[t];
        float* op = args.out + args.ooff[t] + b * nlab;
        for (int c = 0; c < nlab; ++c) op[c] = acc[c] + bc[c];
    }
}

// -------------------------------------------------------------------------
extern "C" void kernel_launch(void* const* d_in, const int* in_sizes, int n_in,
                              void* d_out, int out_size, void* d_ws, size_t ws_size,
                              hipStream_t stream) {
    (void)in_sizes; (void)n_in; (void)out_size; (void)ws_size;

    const float* seq = (const float*)d_in[0];
    const int* rel_idx = (const int*)d_in[1]; const int* rel_len = (const int*)d_in[2];
    const int* nov_idx = (const int*)d_in[3]; const int* nov_len = (const int*)d_in[4];
    const int* dir_idx = (const int*)d_in[5]; const int* dir_len = (const int*)d_in[6];

    const float* Wd[3] = { (const float*)d_in[7],  (const float*)d_in[11], (const float*)d_in[15] };
    const float* bd[3] = { (const float*)d_in[8],  (const float*)d_in[12], (const float*)d_in[16] };
    const float* Wc[3] = { (const float*)d_in[9],  (const float*)d_in[13], (const float*)d_in[17] };
    const float* bc[3] = { (const float*)d_in[10], (const float*)d_in[14], (const float*)d_in[18] };

    float* Pt  = (float*)d_ws;                                        // 3*768*64 f32 (pair-interleaved)
    float* Hdn = (float*)((char*)d_ws + (size_t)3 * Hn * Bn * sizeof(float));

    PoolArgs pa;
    pa.seq = seq;
    pa.idx[0] = rel_idx; pa.idx[1] = nov_idx; pa.idx[2] = dir_idx;
    pa.len[0] = rel_len; pa.len[1] = nov_len; pa.len[2] = dir_len;
    pa.Pt = Pt;
    pool_kernel<<<3 * Bn, 256, 0, stream>>>(pa);

    GemmArgs ga;
    for (int i = 0; i < 3; ++i) { ga.Wd[i] = Wd[i]; ga.bd[i] = bd[i]; }
    ga.Pt = Pt; ga.Hdn = Hdn;
    dense_tanh_wmma<<<18, 256, 0, stream>>>(ga);     // 144 waves = 3 tasks * 48 o-tiles

    ClsArgs ca;
    ca.Hdn = Hdn;
    for (int i = 0; i < 3; ++i) { ca.Wc[i] = Wc[i]; ca.bc[i] = bc[i]; }
    ca.out = (float*)d_out;
    ca.nlab[0] = 9; ca.nlab[1] = 3; ca.nlab[2] = 3;
    ca.ooff[0] = 0; ca.ooff[1] = 576; ca.ooff[2] = 768;
    cls_kernel<<<24, 256, 0, stream>>>(ca);
}